// DetectionLayer_35424890257466
// MI455X (gfx1250) — compile-verified
//
#include <hip/hip_runtime.h>
#include <math.h>
#include <stdint.h>

#define IOU_THR 0.7f
#define KEEP_N  300
#define BS      16
#define FH      100
#define FW      168
#define NA      9
#define NPIX    (FH * FW)          // 16800
#define NTOT    (NA * NPIX)        // 151200
#define TILE    256
#define NMS_T   1024

// ---------------------------------------------------------------------------
// Kernel 1: decode.  Async global->LDS staging of the 6 per-anchor dwords,
// then score = sigmoid(p1-p0), anchor decode, clip.  Also produces per-image
// max "reach" (center offset + half extent, unclipped) for the windowed NMS.
// ---------------------------------------------------------------------------
__global__ __launch_bounds__(TILE) void decode_kernel(
    const float* __restrict__ preds, const float* __restrict__ regs,
    float4* __restrict__ oboxes, float* __restrict__ oscores,
    unsigned* __restrict__ rmax /* [BS][2] float-as-uint, pre-zeroed */) {
  __shared__ float sm[6 * TILE];
  const int tid = threadIdx.x;
  const int b   = blockIdx.x;
  const int n   = blockIdx.y * TILE + tid;
  const int nc  = (n < NTOT) ? n : (NTOT - 1);        // clamp tail lanes
  const unsigned voff = (unsigned)nc * 4u;            // byte offset in plane

  const float* pbase = preds + (size_t)b * 2 * NTOT;
  const float* rbase = regs  + (size_t)b * 4 * NTOT;

  unsigned l0 = (unsigned)(uintptr_t)&sm[0 * TILE + tid];
  unsigned l1 = (unsigned)(uintptr_t)&sm[1 * TILE + tid];
  unsigned l2 = (unsigned)(uintptr_t)&sm[2 * TILE + tid];
  unsigned l3 = (unsigned)(uintptr_t)&sm[3 * TILE + tid];
  unsigned l4 = (unsigned)(uintptr_t)&sm[4 * TILE + tid];
  unsigned l5 = (unsigned)(uintptr_t)&sm[5 * TILE + tid];

  unsigned long long s0 = (unsigned long long)(uintptr_t)(pbase);
  unsigned long long s1 = (unsigned long long)(uintptr_t)(pbase + NTOT);
  unsigned long long s2 = (unsigned long long)(uintptr_t)(rbase);
  unsigned long long s3 = (unsigned long long)(uintptr_t)(rbase + NTOT);
  unsigned long long s4 = (unsigned long long)(uintptr_t)(rbase + 2 * NTOT);
  unsigned long long s5 = (unsigned long long)(uintptr_t)(rbase + 3 * NTOT);

  // CDNA5 async path: per-lane global->LDS dword copies, one counter wait.
  asm volatile("global_load_async_to_lds_b32 %0, %1, %2" :: "v"(l0), "v"(voff), "s"(s0) : "memory");
  asm volatile("global_load_async_to_lds_b32 %0, %1, %2" :: "v"(l1), "v"(voff), "s"(s1) : "memory");
  asm volatile("global_load_async_to_lds_b32 %0, %1, %2" :: "v"(l2), "v"(voff), "s"(s2) : "memory");
  asm volatile("global_load_async_to_lds_b32 %0, %1, %2" :: "v"(l3), "v"(voff), "s"(s3) : "memory");
  asm volatile("global_load_async_to_lds_b32 %0, %1, %2" :: "v"(l4), "v"(voff), "s"(s4) : "memory");
  asm volatile("global_load_async_to_lds_b32 %0, %1, %2" :: "v"(l5), "v"(voff), "s"(s5) : "memory");
  asm volatile("s_wait_asynccnt 0" ::: "memory");
  // ASYNCcnt is per-wave and each lane reads only its own slot: no barrier.

  float p0  = sm[0 * TILE + tid];
  float p1  = sm[1 * TILE + tid];
  float dx  = sm[2 * TILE + tid];
  float dy  = sm[3 * TILE + tid];
  float dwv = sm[4 * TILE + tid];
  float dhv = sm[5 * TILE + tid];

  // anchor geometry (computed for all lanes; tail lanes give finite,
  // conservative values that only enter the max-reach reduction)
  int a   = n / NPIX;
  int pix = n - a * NPIX;
  int ri  = a / 3, si = a - ri * 3;
  float sclv = (si == 0) ? 0.5f : (si == 1 ? 1.0f : 2.0f);
  float ratv = (ri == 0) ? 0.5f : (ri == 1 ? 1.0f : 2.0f);
  float sq = sqrtf(ratv);
  float w  = 16.0f * sclv / sq;   // 16*scale*sqrt(1/ratio)
  float h  = 16.0f * sclv * sq;   // 16*scale*sqrt(ratio)
  int y = pix / FW;
  int x = pix - y * FW;
  float cx = 7.5f + 16.0f * (float)x;
  float cy = 7.5f + 16.0f * (float)y;

  float score = 1.0f / (1.0f + expf(p0 - p1));   // softmax[...,1]

  float pcx = dx * w + cx;
  float pcy = dy * h + cy;
  float pw  = expf(dwv) * w;
  float ph  = expf(dhv) * h;

  // reach of the *unclipped* box from its anchor-cell center (>= 0)
  float reach_x = fabsf(pcx - cx) + 0.5f * pw;
  float reach_y = fabsf(pcy - cy) + 0.5f * ph;
  for (int m = 16; m > 0; m >>= 1) {
    reach_x = fmaxf(reach_x, __shfl_xor(reach_x, m, 32));
    reach_y = fmaxf(reach_y, __shfl_xor(reach_y, m, 32));
  }
  if ((tid & 31) == 0) {
    atomicMax(rmax + 2 * b + 0, __float_as_uint(reach_x));  // non-neg: uint order == float order
    atomicMax(rmax + 2 * b + 1, __float_as_uint(reach_y));
  }

  if (n >= NTOT) return;

  const float Wl = FW * 16.0f, Hl = FH * 16.0f;
  float x1 = fminf(fmaxf(pcx - 0.5f * pw, 0.0f), Wl);
  float y1 = fminf(fmaxf(pcy - 0.5f * ph, 0.0f), Hl);
  float x2 = fminf(fmaxf(pcx + 0.5f * pw, 0.0f), Wl);
  float y2 = fminf(fmaxf(pcy + 0.5f * ph, 0.0f), Hl);

  size_t idx = (size_t)b * NTOT + n;
  oboxes[idx]  = make_float4(x1, y1, x2, y2);
  oscores[idx] = score;
}

// ---------------------------------------------------------------------------
// Kernel 2: greedy NMS, one 1024-thread block (32 wave32s) per image.
//  - argmax: cached per-thread chunk maxima + wave32 shuffle reduction
//    (3 __syncthreads per iteration instead of ~12)
//  - suppression: exact spatial window derived from per-image max reach
//  - LDS dirty flags (owner = i & 1023) invalidate cached chunk maxima
// ---------------------------------------------------------------------------
__global__ __launch_bounds__(NMS_T) void nms_kernel(
    const float4* __restrict__ boxes, float* __restrict__ scores,
    const unsigned* __restrict__ rmax, float* __restrict__ out) {
  __shared__ float s_wval[32];
  __shared__ int   s_widx[32];
  __shared__ int   s_flag[NMS_T];
  __shared__ float s_box[4];
  __shared__ int   s_sel;

  const int tid  = threadIdx.x;
  const int wid  = tid >> 5;
  const int lane = tid & 31;
  const int b    = blockIdx.x;
  const float4* bx = boxes + (size_t)b * NTOT;
  float* sc = scores + (size_t)b * NTOT;
  float* o  = out + (size_t)b * KEEP_N * 5;
  const float Rx = __uint_as_float(rmax[2 * b + 0]);
  const float Ry = __uint_as_float(rmax[2 * b + 1]);

  float best = -INFINITY;
  int   bidx = tid;
  s_flag[tid] = 1;               // force initial scan (own slot: no race)

  for (int k = 0; k < KEEP_N; ++k) {
    if (s_flag[tid]) {           // some index in my chunk was -inf'ed
      s_flag[tid] = 0;
      best = -INFINITY; bidx = tid;
      for (int i = tid; i < NTOT; i += NMS_T) {
        float v = sc[i];
        if (v > best) { best = v; bidx = i; }   // first-max in chunk
      }
    }
    // wave-level argmax (tie -> lowest index), temps keep cache intact
    float rv = best; int ridx = bidx;
    for (int m = 16; m > 0; m >>= 1) {
      float ov = __shfl_xor(rv, m, 32);
      int   oi = __shfl_xor(ridx, m, 32);
      if (ov > rv || (ov == rv && oi < ridx)) { rv = ov; ridx = oi; }
    }
    if (lane == 0) { s_wval[wid] = rv; s_widx[wid] = ridx; }
    __syncthreads();                                   // A
    if (tid < 32) {                                    // wave 0 finishes
      float rv2 = s_wval[tid]; int ri2 = s_widx[tid];
      for (int m = 16; m > 0; m >>= 1) {
        float ov = __shfl_xor(rv2, m, 32);
        int   oi = __shfl_xor(ri2, m, 32);
        if (ov > rv2 || (ov == rv2 && oi < ri2)) { rv2 = ov; ri2 = oi; }
      }
      if (tid == 0) {
        float4 bb = bx[ri2];
        bool fin = isfinite(rv2);
        o[k * 5 + 0] = fin ? bb.x : 0.0f;
        o[k * 5 + 1] = fin ? bb.y : 0.0f;
        o[k * 5 + 2] = fin ? bb.z : 0.0f;
        o[k * 5 + 3] = fin ? bb.w : 0.0f;
        o[k * 5 + 4] = fin ? rv2  : 0.0f;
        sc[ri2] = -INFINITY;
        s_flag[ri2 & (NMS_T - 1)] = 1;  // owner must refresh its cache
        s_box[0] = bb.x; s_box[1] = bb.y; s_box[2] = bb.z; s_box[3] = bb.w;
        s_sel = ri2;
      }
    }
    __syncthreads();                                   // B
    const float sx1 = s_box[0], sy1 = s_box[1], sx2 = s_box[2], sy2 = s_box[3];
    const float sarea = (sx2 - sx1) * (sy2 - sy1);
    // exact window: only cells whose center can yield inter > 0 (+/-1 slack)
    int xlo = (int)floorf((sx1 - Rx - 7.5f) * (1.0f / 16.0f)) - 1;
    int xhi = (int)ceilf ((sx2 + Rx - 7.5f) * (1.0f / 16.0f)) + 1;
    int ylo = (int)floorf((sy1 - Ry - 7.5f) * (1.0f / 16.0f)) - 1;
    int yhi = (int)ceilf ((sy2 + Ry - 7.5f) * (1.0f / 16.0f)) + 1;
    xlo = max(xlo, 0); xhi = min(xhi, FW - 1);
    ylo = max(ylo, 0); yhi = min(yhi, FH - 1);
    const int nx = xhi - xlo + 1, ny = yhi - ylo + 1;
    const int nxy = nx * ny;
    const int cnt = nxy * NA;
    for (int widx = tid; widx < cnt; widx += NMS_T) {
      int a   = widx / nxy;
      int rem = widx - a * nxy;
      int ry  = rem / nx;
      int i   = a * NPIX + (ylo + ry) * FW + (xlo + rem - ry * nx);
      float4 bb = bx[i];
      float iw = fminf(sx2, bb.z) - fmaxf(sx1, bb.x);
      float ih = fminf(sy2, bb.w) - fmaxf(sy1, bb.y);
      float inter = fmaxf(iw, 0.0f) * fmaxf(ih, 0.0f);
      float area = (bb.z - bb.x) * (bb.w - bb.y);
      float iou = inter / (sarea + area - inter + 1e-9f);
      if (iou > IOU_THR) {
        sc[i] = -INFINITY;
        s_flag[i & (NMS_T - 1)] = 1;     // idempotent; read only after barrier
      }
    }
    __syncthreads();                                   // C
  }
}

extern "C" void kernel_launch(void* const* d_in, const int* in_sizes, int n_in,
                              void* d_out, int out_size, void* d_ws, size_t ws_size,
                              hipStream_t stream) {
  (void)in_sizes; (void)n_in; (void)out_size; (void)ws_size;
  const float* preds = (const float*)d_in[0];   // [16, 18, 100, 168]
  const float* regs  = (const float*)d_in[1];   // [16, 36, 100, 168]
  float* out = (float*)d_out;                   // [16, 300, 5]

  // ws layout: boxes [BS*NTOT] float4 | scores [BS*NTOT] float | rmax [BS*2] u32
  float4*   wsBoxes  = (float4*)d_ws;
  float*    wsScores = (float*)((char*)d_ws + (size_t)BS * NTOT * sizeof(float4));
  unsigned* wsRmax   = (unsigned*)(wsScores + (size_t)BS * NTOT);

  hipMemsetAsync(wsRmax, 0, BS * 2 * sizeof(unsigned), stream);
  dim3 g1(BS, (NTOT + TILE - 1) / TILE);
  decode_kernel<<<g1, TILE, 0, stream>>>(preds, regs, wsBoxes, wsScores, wsRmax);
  nms_kernel<<<BS, NMS_T, 0, stream>>>(wsBoxes, wsScores, wsRmax, out);
}